// GCNModel_39505109188896
// MI455X (gfx1250) — compile-verified
//
#include <hip/hip_runtime.h>
#include <math.h>

typedef __attribute__((ext_vector_type(2))) float v2f;
typedef __attribute__((ext_vector_type(8))) float v8f;

#define IN_DIM  128
#define HID_DIM 128
#define OUT_DIM 40
#define OUT_PAD 48
#define LDS_STRIDE 132   // 128 + 4 pad -> 16 distinct LDS banks for m-strided reads

__device__ __forceinline__ void atomAddF(float* p, float v) {
  // fire-and-forget fp32 atomic add; relaxed+agent scope lowers to
  // global_atomic_add_f32 without return (STOREcnt only)
  (void)__hip_atomic_fetch_add(p, v, __ATOMIC_RELAXED, __HIP_MEMORY_SCOPE_AGENT);
}

// ---------------- degree / norm ----------------
__global__ void k_deg_init(float* __restrict__ deg, int n) {
  int i = blockIdx.x * blockDim.x + threadIdx.x;
  if (i < n) deg[i] = 1.0f;  // self loop contributes 1 to every node
}
__global__ void k_deg_edges(const int* __restrict__ row, float* __restrict__ deg, int E) {
  int i = blockIdx.x * blockDim.x + threadIdx.x;
  if (i < E) atomAddF(&deg[row[i]], 1.0f);
}
__global__ void k_rsqrt_inplace(float* __restrict__ deg, int n) {
  int i = blockIdx.x * blockDim.x + threadIdx.x;
  if (i < n) deg[i] = rsqrtf(deg[i]);
}

// ---------------- W2 zero-pad 40 -> 48 cols ----------------
__global__ void k_pad_w2(const float* __restrict__ W2, float* __restrict__ w2p) {
  int i = blockIdx.x * blockDim.x + threadIdx.x;
  if (i < IN_DIM * OUT_PAD) {
    int k = i / OUT_PAD, c = i % OUT_PAD;
    w2p[i] = (c < OUT_DIM) ? W2[k * OUT_DIM + c] : 0.0f;
  }
}

// ---------------- fp32 WMMA GEMM: C[nrows x LDC] = A[nrows x 128] @ B[128 x LDB] -------
// One wave per 16x16 output tile; NW waves cover NW*16 columns; K in steps of 4
// via V_WMMA_F32_16X16X4_F32 (exact fp32). LDB/LDC compile-time so all address
// math strength-reduces (no v_mul_u64 in the epilogue).
template<int NW, int LDB, int LDC>
__global__ __launch_bounds__(NW * 32)
void k_gemm_wmma(const float* __restrict__ A, const float* __restrict__ B,
                 float* __restrict__ C, int nrows) {
  __shared__ float As[16 * LDS_STRIDE];
  const int tid  = threadIdx.x;
  const int lane = tid & 31;
  const int wave = tid >> 5;
  const int m    = lane & 15;   // row within tile (A) / col within tile (B,C)
  const int half = lane >> 4;   // upper/lower half-wave
  const long row0 = (long)blockIdx.x * 16;

  // cooperative load of the 16x128 A slab (float4), zero-fill OOB rows
  for (int i = tid; i < 16 * (IN_DIM / 4); i += NW * 32) {
    int r  = i >> 5;            // 32 float4 per row
    int c4 = i & 31;
    float4 v = make_float4(0.f, 0.f, 0.f, 0.f);
    if (row0 + r < nrows) v = ((const float4*)(A + (row0 + r) * IN_DIM))[c4];
    *((float4*)(As + r * LDS_STRIDE) + c4) = v;
  }
  __syncthreads();

  const int col0 = wave * 16;
  v8f acc = {};
#pragma unroll 4
  for (int k = 0; k < IN_DIM; k += 4) {
    v2f a, b;
    // A 16x4 fp32 layout: lane m, VGPR0 = K = 2*half, VGPR1 = K+1
    a.x = As[m * LDS_STRIDE + k + 2 * half];
    a.y = As[m * LDS_STRIDE + k + 2 * half + 1];
    // B 4x16 fp32 layout (mirrored): lane gives N=m, VGPR v -> K = v + 2*half
    const float* Bp = B + (k + 2 * half) * LDB + col0 + m;
    b.x = Bp[0];
    b.y = Bp[LDB];
    acc = __builtin_amdgcn_wmma_f32_16x16x4_f32(false, a, false, b,
                                                (short)0, acc, false, false);
  }

  // C/D layout: VGPR r -> row r + 8*half, col m
  const int col = col0 + m;
  if (row0 + 16 <= nrows) {                 // fast path: whole tile in range
    if (col < LDC) {
      float* Cp = C + (row0 + 8 * half) * LDC + col;
#pragma unroll
      for (int r = 0; r < 8; ++r) Cp[r * LDC] = acc[r];
    }
  } else {                                  // tail tile (unused when N%16==0)
    if (col < LDC) {
#pragma unroll
      for (int r = 0; r < 8; ++r) {
        long row = row0 + r + 8 * half;
        if (row < nrows) C[row * LDC + col] = acc[r];
      }
    }
  }
}

// ---------------- self-loop init: agg[i,:] = dinv[i]^2 * h[i,:] ----------------
__global__ void k_selfloop128(const float* __restrict__ h, const float* __restrict__ dinv,
                              float* __restrict__ agg, int n) {
  long idx = (long)blockIdx.x * blockDim.x + threadIdx.x;  // over n*32 float4
  if (idx < (long)n * 32) {
    int node = (int)(idx >> 5);
    float s = dinv[node]; s *= s;
    float4 v = ((const float4*)h)[idx];
    ((float4*)agg)[idx] = make_float4(s * v.x, s * v.y, s * v.z, s * v.w);
  }
}
__global__ void k_selfloop40(const float* __restrict__ h2, const float* __restrict__ dinv,
                             float* __restrict__ out, int n) {
  long i = (long)blockIdx.x * blockDim.x + threadIdx.x;  // over n*40
  if (i < (long)n * OUT_DIM) {
    int node = (int)(i / OUT_DIM);
    float s = dinv[node];
    out[i] = s * s * h2[i];
  }
}

// ---------------- edge scatter: one wave per edge ----------------
__global__ void k_edge128(const int* __restrict__ ei, const float* __restrict__ ew,
                          const float* __restrict__ dinv, const float* __restrict__ h,
                          float* __restrict__ agg, int E) {
  int e    = blockIdx.x * 8 + (threadIdx.x >> 5);
  int lane = threadIdx.x & 31;
  if (e >= E) return;
  int r = ei[e], c = ei[E + e];
  float nrm = ew[e] * dinv[r] * dinv[c];
  float4 v = ((const float4*)(h + (long)r * HID_DIM))[lane];  // exactly 1 f4 per lane
  float* dst = agg + (long)c * HID_DIM + lane * 4;
  atomAddF(dst + 0, nrm * v.x);
  atomAddF(dst + 1, nrm * v.y);
  atomAddF(dst + 2, nrm * v.z);
  atomAddF(dst + 3, nrm * v.w);
}
__global__ void k_edge40(const int* __restrict__ ei, const float* __restrict__ ew,
                         const float* __restrict__ dinv, const float* __restrict__ h2,
                         float* __restrict__ out, int E) {
  int e    = blockIdx.x * 8 + (threadIdx.x >> 5);
  int lane = threadIdx.x & 31;
  if (e >= E) return;
  int r = ei[e], c = ei[E + e];
  float nrm = ew[e] * dinv[r] * dinv[c];
  const float* hs = h2 + (long)r * OUT_DIM;
  float* dst = out + (long)c * OUT_DIM;
  atomAddF(dst + lane, nrm * hs[lane]);
  if (lane < OUT_DIM - 32) atomAddF(dst + 32 + lane, nrm * hs[32 + lane]);
}

// ---------------- bias + ReLU ----------------
__global__ void k_relu_bias(const float* __restrict__ agg, const float* __restrict__ b1,
                            float* __restrict__ out, long total) {
  long i = (long)blockIdx.x * blockDim.x + threadIdx.x;
  if (i < total) {
    float v = agg[i] + b1[i & (HID_DIM - 1)];
    out[i] = v > 0.f ? v : 0.f;
  }
}

// ---------------- bias + log_softmax (in place), one wave per node ----------------
__global__ void k_lsm(float* __restrict__ out, const float* __restrict__ b2, int n) {
  int node = blockIdx.x * 8 + (threadIdx.x >> 5);
  int lane = threadIdx.x & 31;
  if (node >= n) return;
  float* p = out + (long)node * OUT_DIM;
  float v0 = p[lane] + b2[lane];
  float v1 = (lane < OUT_DIM - 32) ? (p[32 + lane] + b2[32 + lane]) : -1e30f;
  float mx = fmaxf(v0, v1);
  for (int o = 16; o > 0; o >>= 1) mx = fmaxf(mx, __shfl_xor(mx, o, 32));
  float s = expf(v0 - mx) + ((lane < OUT_DIM - 32) ? expf(v1 - mx) : 0.f);
  for (int o = 16; o > 0; o >>= 1) s += __shfl_xor(s, o, 32);
  float ls = logf(s);
  p[lane] = v0 - mx - ls;
  if (lane < OUT_DIM - 32) p[32 + lane] = v1 - mx - ls;
}

extern "C" void kernel_launch(void* const* d_in, const int* in_sizes, int n_in,
                              void* d_out, int out_size, void* d_ws, size_t ws_size,
                              hipStream_t stream) {
  const float* x  = (const float*)d_in[0];
  const int*   ei = (const int*)d_in[1];
  const float* ew = (const float*)d_in[2];
  const float* W1 = (const float*)d_in[3];
  const float* b1 = (const float*)d_in[4];
  const float* W2 = (const float*)d_in[5];
  const float* b2 = (const float*)d_in[6];
  float* out = (float*)d_out;

  const int N = in_sizes[0] / IN_DIM;
  const int E = in_sizes[2];

  // workspace carve-up (256B aligned): dinv | bufA (h1 / relu) | bufB (agg1 / h2) | W2pad
  char* ws = (char*)d_ws;
  size_t off = 0;
  auto carve = [&](size_t bytes) -> float* {
    float* p = (float*)(ws + off);
    off = (off + bytes + 255) & ~(size_t)255;
    return p;
  };
  float* dinv  = carve((size_t)N * sizeof(float));
  float* bufA  = carve((size_t)N * IN_DIM * sizeof(float));
  float* bufB  = carve((size_t)N * IN_DIM * sizeof(float));
  float* w2pad = carve((size_t)IN_DIM * OUT_PAD * sizeof(float));
  (void)ws_size; (void)n_in; (void)out_size;

  // 1) degree + rsqrt (dinv doubles as deg buffer)
  k_deg_init<<<(N + 255) / 256, 256, 0, stream>>>(dinv, N);
  k_deg_edges<<<(E + 255) / 256, 256, 0, stream>>>(ei, dinv, E);
  k_rsqrt_inplace<<<(N + 255) / 256, 256, 0, stream>>>(dinv, N);

  // 2) zero-padded W2 (128x48)
  k_pad_w2<<<(IN_DIM * OUT_PAD + 255) / 256, 256, 0, stream>>>(W2, w2pad);

  const int rowTiles = (N + 15) / 16;

  // 3) h1 = X @ W1  (fp32 WMMA, 8 waves cover 128 cols)
  k_gemm_wmma<8, HID_DIM, HID_DIM><<<rowTiles, 256, 0, stream>>>(x, W1, bufA, N);

  // 4) layer-1 aggregation: self loops then edge scatter
  k_selfloop128<<<(int)(((long)N * 32 + 255) / 256), 256, 0, stream>>>(bufA, dinv, bufB, N);
  k_edge128<<<(E + 7) / 8, 256, 0, stream>>>(ei, ew, dinv, bufA, bufB, E);

  // 5) bias + ReLU -> bufA
  k_relu_bias<<<(int)(((long)N * HID_DIM + 255) / 256), 256, 0, stream>>>(
      bufB, b1, bufA, (long)N * HID_DIM);

  // 6) h2 = relu @ W2pad (3 waves cover 48 padded cols; store 40 real) -> bufB
  k_gemm_wmma<3, OUT_PAD, OUT_DIM><<<rowTiles, 96, 0, stream>>>(bufA, w2pad, bufB, N);

  // 7) layer-2 aggregation directly into d_out
  k_selfloop40<<<(int)(((long)N * OUT_DIM + 255) / 256), 256, 0, stream>>>(bufB, dinv, out, N);
  k_edge40<<<(E + 7) / 8, 256, 0, stream>>>(ei, ew, dinv, bufB, out, E);

  // 8) bias + log_softmax in place
  k_lsm<<<(N + 7) / 8, 256, 0, stream>>>(out, b2, N);
}